// GAT_58179626992389
// MI455X (gfx1250) — compile-verified
//
#include <hip/hip_runtime.h>
#include <math.h>
#include <stdint.h>

typedef __attribute__((ext_vector_type(2))) float v2f;
typedef __attribute__((ext_vector_type(8))) float v8f;

#define NFEAT 512
#define HID   64
#define HEADS 8
#define FH    8
#define NCLS  40
#define NCP   48      // padded class columns (3 WMMA tiles)
#define NEGS  0.2f

// ---------------------------------------------------------------- utilities
__global__ void fill_kernel(float* __restrict__ p, float v, int n) {
  int i = blockIdx.x * blockDim.x + threadIdx.x;
  if (i < n) p[i] = v;
}

__device__ __forceinline__ float lrelu(float v) { return v > 0.f ? v : NEGS * v; }

// Async global->LDS B128 copy (CDNA5, ASYNCcnt-tracked)
__device__ __forceinline__ void async_copy_b128(uint32_t lds_byte_off, uint64_t gaddr) {
  asm volatile("global_load_async_to_lds_b128 %0, %1, off"
               :: "v"(lds_byte_off), "v"(gaddr) : "memory");
}
__device__ __forceinline__ void async_wait0() {
  asm volatile("s_wait_asynccnt 0" ::: "memory");
}

// ---------------------------------------------------------------- weight pre-pack
// Wp[(k>>1)][c] = float2{ W[k][c], W[k+1][c] }  -> one ds_load_b64 per WMMA B operand
__global__ void pack1_kernel(const float* __restrict__ W1, float* __restrict__ Wp) {
  int i = blockIdx.x * blockDim.x + threadIdx.x;      // over 512*64
  if (i >= NFEAT * HID) return;
  const int k = i / HID, c = i % HID;
  Wp[((size_t)(k >> 1) * HID + c) * 2 + (k & 1)] = W1[i];
}

__global__ void pack2_kernel(const float* __restrict__ W2, float* __restrict__ Wp) {
  int i = blockIdx.x * blockDim.x + threadIdx.x;      // over 64*48 (padded)
  if (i >= HID * NCP) return;
  const int k = i / NCP, c = i % NCP;
  Wp[((size_t)(k >> 1) * NCP + c) * 2 + (k & 1)] = (c < NCLS) ? W2[k * NCLS + c] : 0.f;
}

// ---------------------------------------------------------------- GEMM 1
// h1[N,64] = x[N,512] @ W1[512,64], fp32 WMMA 16x16x4.
// Wave: 16(M) x 64(N) tile. Packed W chunks DMA'd to LDS asynchronously.
__global__ __launch_bounds__(256) void gemm1_kernel(const float* __restrict__ x,
                                                    const float* __restrict__ Wp1,
                                                    float* __restrict__ h1,
                                                    int nNodes) {
  const int KC = 128;                                  // k rows per chunk
  __shared__ __align__(16) float lw[(KC / 2) * HID * 2];  // 32 KB packed float2
  const int lane = threadIdx.x & 31;
  const int wave = threadIdx.x >> 5;
  const int l    = lane & 15;
  const int half = lane >> 4;                          // 0: K,K+1  1: K+2,K+3
  const int mbase = blockIdx.x * 128 + wave * 16;
  int row = mbase + l;
  if (row >= nNodes) row = nNodes - 1;                 // keep EXEC all-ones for WMMA

  const uint32_t lbase = (uint32_t)(uintptr_t)(void*)lw;
  const v2f* __restrict__ bp = (const v2f*)lw;

  v8f acc0 = {}, acc1 = {}, acc2 = {}, acc3 = {};
  for (int kc = 0; kc < NFEAT; kc += KC) {
    __syncthreads();
    {
      const uint64_t gsrc = (uint64_t)(uintptr_t)(Wp1 + (size_t)(kc / 2) * HID * 2);
      #pragma unroll
      for (int it = 0; it < 8; ++it) {                 // 8 * 256 * 16B = 32 KB
        const uint32_t off = (uint32_t)(it * 256 + threadIdx.x) * 16u;
        async_copy_b128(lbase + off, gsrc + off);
      }
      async_wait0();
    }
    __syncthreads();
    #pragma unroll 4
    for (int kk = 0; kk < KC; kk += 4) {
      const int k  = kk + 2 * half;
      const int k2 = (kk >> 1) + half;                 // packed pair row
      v2f a = *(const v2f*)&x[(size_t)row * NFEAT + kc + k];   // A[M=l][k..k+1]
      v2f b0 = bp[(size_t)k2 * HID +  0 + l];
      v2f b1 = bp[(size_t)k2 * HID + 16 + l];
      v2f b2 = bp[(size_t)k2 * HID + 32 + l];
      v2f b3 = bp[(size_t)k2 * HID + 48 + l];
      acc0 = __builtin_amdgcn_wmma_f32_16x16x4_f32(false, a, false, b0, (short)0, acc0, false, false);
      acc1 = __builtin_amdgcn_wmma_f32_16x16x4_f32(false, a, false, b1, (short)0, acc1, false, false);
      acc2 = __builtin_amdgcn_wmma_f32_16x16x4_f32(false, a, false, b2, (short)0, acc2, false, false);
      acc3 = __builtin_amdgcn_wmma_f32_16x16x4_f32(false, a, false, b3, (short)0, acc3, false, false);
    }
  }
  #pragma unroll
  for (int r = 0; r < 8; ++r) {                        // C: VGPR r -> rows r / r+8
    const int node = mbase + r + 8 * half;
    if (node < nNodes) {
      h1[(size_t)node * HID +  0 + l] = acc0[r];
      h1[(size_t)node * HID + 16 + l] = acc1[r];
      h1[(size_t)node * HID + 32 + l] = acc2[r];
      h1[(size_t)node * HID + 48 + l] = acc3[r];
    }
  }
}

// ---------------------------------------------------------------- GEMM 2
// t2[N,40] = h2[N,64] @ W2[64,40]; columns padded 40->48 (3 WMMA tiles).
__global__ __launch_bounds__(256) void gemm2_kernel(const float* __restrict__ h2,
                                                    const float* __restrict__ Wp2,
                                                    float* __restrict__ t2,
                                                    int nNodes) {
  __shared__ __align__(16) float lw[(HID / 2) * NCP * 2];  // 12 KB packed float2
  const int lane = threadIdx.x & 31;
  const int wave = threadIdx.x >> 5;
  const int l    = lane & 15;
  const int half = lane >> 4;
  const int mbase = blockIdx.x * 128 + wave * 16;
  int row = mbase + l;
  if (row >= nNodes) row = nNodes - 1;

  const uint32_t lbase = (uint32_t)(uintptr_t)(void*)lw;
  const uint64_t gsrc  = (uint64_t)(uintptr_t)Wp2;
  #pragma unroll
  for (int it = 0; it < 3; ++it) {                     // 3 * 256 * 16B = 12 KB
    const uint32_t off = (uint32_t)(it * 256 + threadIdx.x) * 16u;
    async_copy_b128(lbase + off, gsrc + off);
  }
  async_wait0();
  __syncthreads();

  const v2f* __restrict__ bp = (const v2f*)lw;
  v8f acc0 = {}, acc1 = {}, acc2 = {};
  #pragma unroll 4
  for (int kk = 0; kk < HID; kk += 4) {
    const int k  = kk + 2 * half;
    const int k2 = (kk >> 1) + half;
    v2f a = *(const v2f*)&h2[(size_t)row * HID + k];
    v2f b0 = bp[(size_t)k2 * NCP +  0 + l];
    v2f b1 = bp[(size_t)k2 * NCP + 16 + l];
    v2f b2 = bp[(size_t)k2 * NCP + 32 + l];
    acc0 = __builtin_amdgcn_wmma_f32_16x16x4_f32(false, a, false, b0, (short)0, acc0, false, false);
    acc1 = __builtin_amdgcn_wmma_f32_16x16x4_f32(false, a, false, b1, (short)0, acc1, false, false);
    acc2 = __builtin_amdgcn_wmma_f32_16x16x4_f32(false, a, false, b2, (short)0, acc2, false, false);
  }
  #pragma unroll
  for (int r = 0; r < 8; ++r) {
    const int node = mbase + r + 8 * half;
    if (node < nNodes) {
      t2[(size_t)node * NCLS +  0 + l] = acc0[r];
      t2[(size_t)node * NCLS + 16 + l] = acc1[r];
      if (32 + l < NCLS) t2[(size_t)node * NCLS + 32 + l] = acc2[r];
    }
  }
}

// ---------------------------------------------------------------- attention logits
__global__ void att1_kernel(const float* __restrict__ h1, const float* __restrict__ att_s,
                            const float* __restrict__ att_d, float* __restrict__ as,
                            float* __restrict__ ad, int nNodes) {
  int n = blockIdx.x * blockDim.x + threadIdx.x;
  if (n >= nNodes) return;
  #pragma unroll
  for (int hh = 0; hh < HEADS; ++hh) {
    float s = 0.f, d = 0.f;
    #pragma unroll
    for (int f = 0; f < FH; ++f) {
      const float hv = h1[(size_t)n * HID + hh * FH + f];
      s += hv * att_s[hh * FH + f];
      d += hv * att_d[hh * FH + f];
    }
    as[(size_t)n * HEADS + hh] = s;
    ad[(size_t)n * HEADS + hh] = d;
  }
}

__global__ void att2_kernel(const float* __restrict__ t2, const float* __restrict__ att_s,
                            const float* __restrict__ att_d, float* __restrict__ as,
                            float* __restrict__ ad, int nNodes) {
  int n = blockIdx.x * blockDim.x + threadIdx.x;
  if (n >= nNodes) return;
  float s = 0.f, d = 0.f;
  #pragma unroll 8
  for (int c = 0; c < NCLS; ++c) {
    const float hv = t2[(size_t)n * NCLS + c];
    s += hv * att_s[c];
    d += hv * att_d[c];
  }
  as[n] = s;
  ad[n] = d;
}

// ---------------------------------------------------------------- edge helpers
__device__ __forceinline__ void edge_sd(const int* __restrict__ ei, int Eo, int e,
                                        int& s, int& d) {
  if (e < Eo) { s = ei[e]; d = ei[Eo + e]; }
  else        { s = e - Eo; d = s; }                   // self loops appended
}

// ---------------------------------------------------------------- layer-1 edges
__global__ void edge_max1_kernel(const int* __restrict__ ei, int Eo, int Et,
                                 const float* __restrict__ as, const float* __restrict__ ad,
                                 float* __restrict__ m) {
  int e = blockIdx.x * blockDim.x + threadIdx.x;
  if (e >= Et) return;
  int s, d; edge_sd(ei, Eo, e, s, d);
  #pragma unroll
  for (int hh = 0; hh < HEADS; ++hh) {
    const float v = lrelu(as[(size_t)s * HEADS + hh] + ad[(size_t)d * HEADS + hh]);
    atomicMax(&m[(size_t)d * HEADS + hh], v);
  }
}

__global__ void edge_sum1_kernel(const int* __restrict__ ei, int Eo, int Et,
                                 const float* __restrict__ as, const float* __restrict__ ad,
                                 const float* __restrict__ m, float* __restrict__ dn) {
  int e = blockIdx.x * blockDim.x + threadIdx.x;
  if (e >= Et) return;
  int s, d; edge_sd(ei, Eo, e, s, d);
  #pragma unroll
  for (int hh = 0; hh < HEADS; ++hh) {
    const float v = lrelu(as[(size_t)s * HEADS + hh] + ad[(size_t)d * HEADS + hh]);
    atomicAdd(&dn[(size_t)d * HEADS + hh], expf(v - m[(size_t)d * HEADS + hh]));
  }
}

__global__ void edge_agg1_kernel(const int* __restrict__ ei, int Eo, int Et,
                                 const float* __restrict__ as, const float* __restrict__ ad,
                                 const float* __restrict__ m, const float* __restrict__ dn,
                                 const float* __restrict__ h1, float* __restrict__ agg) {
  int e = blockIdx.x * blockDim.x + threadIdx.x;
  if (e >= Et) return;
  int s, d; edge_sd(ei, Eo, e, s, d);
  float alpha[HEADS];
  #pragma unroll
  for (int hh = 0; hh < HEADS; ++hh) {
    const float v = lrelu(as[(size_t)s * HEADS + hh] + ad[(size_t)d * HEADS + hh]);
    alpha[hh] = expf(v - m[(size_t)d * HEADS + hh]) / (dn[(size_t)d * HEADS + hh] + 1e-16f);
  }
  #pragma unroll 8
  for (int f = 0; f < HID; ++f)
    atomicAdd(&agg[(size_t)d * HID + f], h1[(size_t)s * HID + f] * alpha[f >> 3]);
}

// ELU(agg + b1) in place -> becomes h2
__global__ void node1_kernel(float* __restrict__ agg, const float* __restrict__ b1, int n) {
  int i = blockIdx.x * blockDim.x + threadIdx.x;
  if (i >= n) return;
  const float v = agg[i] + b1[i & (HID - 1)];
  agg[i] = v > 0.f ? v : (expf(v) - 1.f);
}

// ---------------------------------------------------------------- layer-2 edges (heads=1)
__global__ void edge_max2_kernel(const int* __restrict__ ei, int Eo, int Et,
                                 const float* __restrict__ as, const float* __restrict__ ad,
                                 float* __restrict__ m) {
  int e = blockIdx.x * blockDim.x + threadIdx.x;
  if (e >= Et) return;
  int s, d; edge_sd(ei, Eo, e, s, d);
  atomicMax(&m[d], lrelu(as[s] + ad[d]));
}

__global__ void edge_sum2_kernel(const int* __restrict__ ei, int Eo, int Et,
                                 const float* __restrict__ as, const float* __restrict__ ad,
                                 const float* __restrict__ m, float* __restrict__ dn) {
  int e = blockIdx.x * blockDim.x + threadIdx.x;
  if (e >= Et) return;
  int s, d; edge_sd(ei, Eo, e, s, d);
  atomicAdd(&dn[d], expf(lrelu(as[s] + ad[d]) - m[d]));
}

__global__ void edge_agg2_kernel(const int* __restrict__ ei, int Eo, int Et,
                                 const float* __restrict__ as, const float* __restrict__ ad,
                                 const float* __restrict__ m, const float* __restrict__ dn,
                                 const float* __restrict__ t2, float* __restrict__ out) {
  int e = blockIdx.x * blockDim.x + threadIdx.x;
  if (e >= Et) return;
  int s, d; edge_sd(ei, Eo, e, s, d);
  const float alpha = expf(lrelu(as[s] + ad[d]) - m[d]) / (dn[d] + 1e-16f);
  #pragma unroll 8
  for (int c = 0; c < NCLS; ++c)
    atomicAdd(&out[(size_t)d * NCLS + c], t2[(size_t)s * NCLS + c] * alpha);
}

__global__ void bias2_kernel(float* __restrict__ out, const float* __restrict__ b2, int n) {
  int i = blockIdx.x * blockDim.x + threadIdx.x;
  if (i < n) out[i] += b2[i % NCLS];
}

// ---------------------------------------------------------------- launcher
extern "C" void kernel_launch(void* const* d_in, const int* in_sizes, int n_in,
                              void* d_out, int out_size, void* d_ws, size_t ws_size,
                              hipStream_t stream) {
  (void)n_in; (void)out_size; (void)ws_size;
  const float* x    = (const float*)d_in[0];
  const int*   ei   = (const int*)  d_in[1];
  const float* W1   = (const float*)d_in[2];
  const float* as1w = (const float*)d_in[3];
  const float* ad1w = (const float*)d_in[4];
  const float* b1   = (const float*)d_in[5];
  const float* W2   = (const float*)d_in[6];
  const float* as2w = (const float*)d_in[7];
  const float* ad2w = (const float*)d_in[8];
  const float* b2   = (const float*)d_in[9];
  float* out = (float*)d_out;

  const int N  = in_sizes[0] / NFEAT;   // 100000
  const int Eo = in_sizes[1] / 2;       // 1600000
  const int Et = Eo + N;                // + self loops

  float* ws  = (float*)d_ws;
  float* h1  = ws;                           // N*64
  float* ag1 = h1  + (size_t)N * HID;        // N*64 (becomes h2)
  float* as1 = ag1 + (size_t)N * HID;        // N*8
  float* ad1 = as1 + (size_t)N * HEADS;      // N*8
  float* m1  = ad1 + (size_t)N * HEADS;      // N*8
  float* dn1 = m1  + (size_t)N * HEADS;      // N*8
  float* t2  = dn1 + (size_t)N * HEADS;      // N*40
  float* as2 = t2  + (size_t)N * NCLS;       // N
  float* ad2 = as2 + N;                      // N
  float* m2  = ad2 + N;                      // N
  float* dn2 = m2  + N;                      // N
  float* Wp1 = dn2 + N;                      // 512*64 packed pairs
  float* Wp2 = Wp1 + (size_t)NFEAT * HID;    // 64*48 packed pairs (padded)

  auto nb = [](long n, int b) { return (unsigned)((n + b - 1) / b); };

  fill_kernel<<<nb((long)N * HID, 256), 256, 0, stream>>>(ag1, 0.f, N * HID);
  fill_kernel<<<nb((long)N * HEADS, 256), 256, 0, stream>>>(m1, -3.0e38f, N * HEADS);
  fill_kernel<<<nb((long)N * HEADS, 256), 256, 0, stream>>>(dn1, 0.f, N * HEADS);
  fill_kernel<<<nb(N, 256), 256, 0, stream>>>(m2, -3.0e38f, N);
  fill_kernel<<<nb(N, 256), 256, 0, stream>>>(dn2, 0.f, N);
  fill_kernel<<<nb((long)N * NCLS, 256), 256, 0, stream>>>(out, 0.f, N * NCLS);

  pack1_kernel<<<nb(NFEAT * HID, 256), 256, 0, stream>>>(W1, Wp1);
  pack2_kernel<<<nb(HID * NCP, 256), 256, 0, stream>>>(W2, Wp2);

  gemm1_kernel<<<nb(N, 128), 256, 0, stream>>>(x, Wp1, h1, N);
  att1_kernel<<<nb(N, 256), 256, 0, stream>>>(h1, as1w, ad1w, as1, ad1, N);
  edge_max1_kernel<<<nb(Et, 256), 256, 0, stream>>>(ei, Eo, Et, as1, ad1, m1);
  edge_sum1_kernel<<<nb(Et, 256), 256, 0, stream>>>(ei, Eo, Et, as1, ad1, m1, dn1);
  edge_agg1_kernel<<<nb(Et, 256), 256, 0, stream>>>(ei, Eo, Et, as1, ad1, m1, dn1, h1, ag1);
  node1_kernel<<<nb((long)N * HID, 256), 256, 0, stream>>>(ag1, b1, N * HID);

  gemm2_kernel<<<nb(N, 128), 256, 0, stream>>>(ag1, Wp2, t2, N);
  att2_kernel<<<nb(N, 256), 256, 0, stream>>>(t2, as2w, ad2w, as2, ad2, N);
  edge_max2_kernel<<<nb(Et, 256), 256, 0, stream>>>(ei, Eo, Et, as2, ad2, m2);
  edge_sum2_kernel<<<nb(Et, 256), 256, 0, stream>>>(ei, Eo, Et, as2, ad2, m2, dn2);
  edge_agg2_kernel<<<nb(Et, 256), 256, 0, stream>>>(ei, Eo, Et, as2, ad2, m2, dn2, t2, out);
  bias2_kernel<<<nb((long)N * NCLS, 256), 256, 0, stream>>>(out, b2, N * NCLS);
}